// VoxelPredRefiner_63591285785174
// MI455X (gfx1250) — compile-verified
//
#include <hip/hip_runtime.h>
#include <hip/hip_bf16.h>
#include <stdint.h>

// Problem constants (from reference)
#define NPTS      500000
#define PF        17
#define CBEV      256
#define HW        180
#define IN0_DIM   272   // 16 point feats + 256 bev channels
#define IN1_DIM   64
#define NC        20
#define NC_PAD    32    // classes padded to 2 WMMA N-tiles
#define BN_EPS    1e-3f

#define XS_STRIDE 274   // padded LDS stride for 272-wide tiles (bank-conflict free, 8B aligned)
#define HS_STRIDE 68    // padded LDS stride for 64-wide h / W2 tiles (banks 4n+k, conflict free)
#define ROWS_PER_BLOCK 64
#define THREADS   128   // 4 wave32 waves

typedef float v2f __attribute__((ext_vector_type(2)));
typedef float v8f __attribute__((ext_vector_type(8)));

static __device__ __forceinline__ v8f v8f_zero() {
  v8f z;
#pragma unroll
  for (int i = 0; i < 8; ++i) z[i] = 0.0f;
  return z;
}

// D = A(16x4 f32) * B(4x16 f32) + C(16x16 f32), fp32 WMMA (exact vs fp32 reference)
static __device__ __forceinline__ v8f wmma_f32(v2f a, v2f b, v8f c) {
  return __builtin_amdgcn_wmma_f32_16x16x4_f32(
      /*neg_a=*/false, a, /*neg_b=*/false, b,
      /*c_mod=*/(short)0, c, /*reuse_a=*/false, /*reuse_b=*/false);
}

// ---------------------------------------------------------------------------
// Cooperative staging of one 64-row x-tile (concat[pf[:,1:17], bev gather])
// and the full W1 (64x272) into LDS.
// ---------------------------------------------------------------------------
static __device__ __forceinline__ void stage_tiles(
    const int* __restrict__ coor, const float* __restrict__ pfeat,
    const float* __restrict__ bev, const float* __restrict__ W1,
    float* xs, float* w1s, int* bevbase, int row0) {
  const int tid = threadIdx.x;
  // Per-row gather base index into bev_feature (B,C,H,W): computed once per row.
  for (int r = tid; r < ROWS_PER_BLOCK; r += THREADS) {
    int row = row0 + r;
    int base = 0;
    if (row < NPTS) {
      int b = coor[row * 4 + 0];
      int y = coor[row * 4 + 2] >> 3; y = (y > HW - 1) ? (HW - 1) : y;
      int x = coor[row * 4 + 3] >> 3; x = (x > HW - 1) ? (HW - 1) : x;
      base = ((b * CBEV) * HW + y) * HW + x;
    }
    bevbase[r] = base;
  }
  __syncthreads();
  // x tile: rows beyond N are zero-filled (excluded from stats / stores later).
  for (int idx = tid; idx < ROWS_PER_BLOCK * IN0_DIM; idx += THREADS) {
    int r = idx / IN0_DIM;
    int c = idx - r * IN0_DIM;
    int row = row0 + r;
    float v = 0.0f;
    if (row < NPTS) {
      v = (c < 16) ? pfeat[row * PF + 1 + c]
                   : bev[bevbase[r] + (c - 16) * (HW * HW)];
    }
    xs[r * XS_STRIDE + c] = v;
  }
  // W1 tile (entire 64x272 weight, coalesced load, reused by all 4 waves)
  for (int idx = tid; idx < IN1_DIM * IN0_DIM; idx += THREADS) {
    int r = idx / IN0_DIM;
    int c = idx - r * IN0_DIM;
    w1s[r * XS_STRIDE + c] = W1[idx];
  }
  __syncthreads();
}

// ---------------------------------------------------------------------------
// Per-wave GEMM1: h-tile(16x64) = x-tile(16x272) @ W1^T, as 68 K-steps x 4
// N-tiles of v_wmma_f32_16x16x4_f32.  A/B fragments per ISA layout:
//   lane = 16*kh + m ; A: row m, K = k0+2*kh+{0,1}; B: N = m, same K pair.
// ---------------------------------------------------------------------------
static __device__ __forceinline__ void compute_h_tile(
    const float* xs_wave, const float* w1s, int m, int kh, v8f acc[4]) {
  const float* arow = xs_wave + m * XS_STRIDE + 2 * kh;
#pragma unroll 2
  for (int k0 = 0; k0 < IN0_DIM; k0 += 4) {
    v2f a = *(const v2f*)(arow + k0);  // ds_load_b64, 8B aligned
#pragma unroll
    for (int t = 0; t < 4; ++t) {
      const float* wrow = w1s + (t * 16 + m) * XS_STRIDE + 2 * kh + k0;
      v2f b = *(const v2f*)(wrow);
      acc[t] = wmma_f32(a, b, acc[t]);
    }
  }
}

// ---------------------------------------------------------------------------
// Shared LDS staging for GEMM2 weights: zero-padded W2 (32x64) + b2 (padded)
// + optional scale/shift.  Removes all divergent-EXEC loads from the K-loop.
// ---------------------------------------------------------------------------
static __device__ __forceinline__ void stage_w2(
    const float* __restrict__ W2, const float* __restrict__ b2,
    float* w2s, float* b2s) {
  const int tid = threadIdx.x;
  for (int idx = tid; idx < NC_PAD * IN1_DIM; idx += THREADS) {
    int n = idx >> 6;          // / IN1_DIM
    int k = idx & (IN1_DIM - 1);
    w2s[n * HS_STRIDE + k] = (n < NC) ? W2[n * IN1_DIM + k] : 0.0f;
  }
  if (tid < NC_PAD) b2s[tid] = (tid < NC) ? b2[tid] : 0.0f;
}

// ---------------------------------------------------------------------------
// Per-wave GEMM2 from LDS: out-tile(16x32) = A(16x64) @ W2s^T + b2.
// NORM: apply relu(a*scale+shift) on the A fragments (pass 2a);
// pass 2b pre-applies normalization when writing its h tile to LDS.
// ---------------------------------------------------------------------------
template <bool NORM>
static __device__ __forceinline__ void gemm2_lds(
    const float* hrow, const float* w2s, const float* ssc, const float* ssh,
    const float* b2s, float* __restrict__ out, int row0_wave, int m, int kh) {
  v8f acc2[2];
  acc2[0] = v8f_zero(); acc2[1] = v8f_zero();
#pragma unroll
  for (int k0 = 0; k0 < IN1_DIM; k0 += 4) {
    const int ka = k0 + 2 * kh;
    v2f a = *(const v2f*)(hrow + ka);
    if (NORM) {
      v2f sc = *(const v2f*)(ssc + ka);
      v2f sh = *(const v2f*)(ssh + ka);
      a = a * sc + sh;
      a[0] = fmaxf(a[0], 0.0f);
      a[1] = fmaxf(a[1], 0.0f);
    }
#pragma unroll
    for (int t = 0; t < 2; ++t) {
      v2f b = *(const v2f*)(w2s + (t * 16 + m) * HS_STRIDE + ka);
      acc2[t] = wmma_f32(a, b, acc2[t]);
    }
  }
  const int rowbase = row0_wave + kh * 8;
#pragma unroll
  for (int t = 0; t < 2; ++t) {
    const int col = t * 16 + m;
    if (col < NC) {
      const float bb = b2s[col];
#pragma unroll
      for (int v = 0; v < 8; ++v) {
        int row = rowbase + v;
        if (row < NPTS) out[(size_t)row * NC + col] = acc2[t][v] + bb;
      }
    }
  }
}

// ---------------------------------------------------------------------------
// Pass 1: GEMM1 (+ optional h store to workspace) + BN sum/sumsq reduction.
// NOTE: b1 is omitted everywhere — a constant per-feature bias cancels exactly
// in (h - mean(h)) inside BatchNorm, so the final output is unchanged.
// ---------------------------------------------------------------------------
template <bool STORE_H>
__global__ void __launch_bounds__(THREADS)
vpr_gemm1_stats_kernel(const int* __restrict__ coor,
                       const float* __restrict__ pfeat,
                       const float* __restrict__ bev,
                       const float* __restrict__ W1,
                       float* __restrict__ hws,
                       float* __restrict__ gsum,
                       float* __restrict__ gsumsq) {
  extern __shared__ char smem_raw[];
  float* xs      = (float*)smem_raw;                 // 64 x 274
  float* w1s     = xs + ROWS_PER_BLOCK * XS_STRIDE;  // 64 x 274
  int*   bevbase = (int*)(w1s + IN1_DIM * XS_STRIDE);
  float* ssum    = (float*)(bevbase + ROWS_PER_BLOCK);  // 64
  float* ssq     = ssum + IN1_DIM;                      // 64

  const int tid = threadIdx.x;
  if (tid < 128) ssum[tid] = 0.0f;  // zeros ssum[0..63] and ssq[0..63]

  const int row0 = blockIdx.x * ROWS_PER_BLOCK;
  stage_tiles(coor, pfeat, bev, W1, xs, w1s, bevbase, row0);

  const int wave = tid >> 5, lane = tid & 31;
  const int kh = lane >> 4, m = lane & 15;
  const float* xs_wave = xs + wave * 16 * XS_STRIDE;

  v8f acc[4];
#pragma unroll
  for (int t = 0; t < 4; ++t) acc[t] = v8f_zero();
  compute_h_tile(xs_wave, w1s, m, kh, acc);

  // D layout: lane (16*kh+m), VGPR v -> h[row0+wave*16+v+8*kh][t*16+m]
  const int rowbase = row0 + wave * 16 + kh * 8;
  if (STORE_H) {
#pragma unroll
    for (int t = 0; t < 4; ++t)
#pragma unroll
      for (int v = 0; v < 8; ++v) {
        int row = rowbase + v;
        if (row < NPTS) hws[(size_t)row * IN1_DIM + t * 16 + m] = acc[t][v];
      }
  }

  // Per-feature sum / sumsq over valid rows of this tile.
#pragma unroll
  for (int t = 0; t < 4; ++t) {
    float s = 0.0f, q = 0.0f;
#pragma unroll
    for (int v = 0; v < 8; ++v) {
      int row = rowbase + v;
      if (row < NPTS) { float x = acc[t][v]; s += x; q += x * x; }
    }
    s += __shfl_xor(s, 16, 32);  // combine the two 8-row halves (wave32)
    q += __shfl_xor(q, 16, 32);
    if (kh == 0) {
      atomicAdd(&ssum[t * 16 + m], s);
      atomicAdd(&ssq[t * 16 + m], q);
    }
  }
  __syncthreads();
  if (tid < IN1_DIM)            atomicAdd(&gsum[tid], ssum[tid]);
  else if (tid < 2 * IN1_DIM)   atomicAdd(&gsumsq[tid - IN1_DIM], ssq[tid - IN1_DIM]);
}

// ---------------------------------------------------------------------------
// Fold BN stats + gamma/beta into per-feature scale/shift.
// ---------------------------------------------------------------------------
__global__ void vpr_bn_finalize_kernel(const float* __restrict__ gsum,
                                       const float* __restrict__ gsumsq,
                                       const float* __restrict__ gamma,
                                       const float* __restrict__ beta,
                                       float* __restrict__ scale,
                                       float* __restrict__ shift) {
  int i = threadIdx.x;
  if (i < IN1_DIM) {
    const float inv = 1.0f / (float)NPTS;
    float mean = gsum[i] * inv;
    float var  = gsumsq[i] * inv - mean * mean;
    float sc   = gamma[i] * rsqrtf(var + BN_EPS);
    scale[i] = sc;
    shift[i] = beta[i] - mean * sc;
  }
}

// ---------------------------------------------------------------------------
// Pass 2a: h is in workspace; stage h/W2/scale/shift/b2 into LDS (coalesced),
// then WMMA GEMM2 with on-the-fly normalize+ReLU on the A fragments.
// ---------------------------------------------------------------------------
__global__ void __launch_bounds__(THREADS)
vpr_gemm2_from_h_kernel(const float* __restrict__ hws,
                        const float* __restrict__ scale,
                        const float* __restrict__ shift,
                        const float* __restrict__ W2,
                        const float* __restrict__ b2,
                        float* __restrict__ out) {
  extern __shared__ char smem_raw[];
  float* hs  = (float*)smem_raw;                // 64 x 68
  float* w2s = hs + ROWS_PER_BLOCK * HS_STRIDE; // 32 x 68
  float* ssc = w2s + NC_PAD * HS_STRIDE;        // 64
  float* ssh = ssc + IN1_DIM;                   // 64
  float* b2s = ssh + IN1_DIM;                   // 32

  const int tid = threadIdx.x;
  const int row0 = blockIdx.x * ROWS_PER_BLOCK;

  // Coalesced h-tile stage (rows are contiguous 64-float runs in hws).
  for (int idx = tid; idx < ROWS_PER_BLOCK * IN1_DIM; idx += THREADS) {
    int r = idx >> 6;
    int c = idx & (IN1_DIM - 1);
    int row = row0 + r;
    hs[r * HS_STRIDE + c] = (row < NPTS) ? hws[(size_t)row * IN1_DIM + c] : 0.0f;
  }
  stage_w2(W2, b2, w2s, b2s);
  if (tid < IN1_DIM)            ssc[tid] = scale[tid];
  else if (tid < 2 * IN1_DIM)   ssh[tid - IN1_DIM] = shift[tid - IN1_DIM];
  __syncthreads();

  const int wave = tid >> 5, lane = tid & 31;
  const int kh = lane >> 4, m = lane & 15;
  const float* hrow = hs + (wave * 16 + m) * HS_STRIDE;
  gemm2_lds<true>(hrow, w2s, ssc, ssh, b2s, out, row0 + wave * 16, m, kh);
}

// ---------------------------------------------------------------------------
// Pass 2b: workspace too small for h -> recompute GEMM1 (bev is L2-resident,
// so the regather is cheap), normalize in-register, re-stripe D->A via LDS.
// ---------------------------------------------------------------------------
__global__ void __launch_bounds__(THREADS)
vpr_gemm2_recompute_kernel(const int* __restrict__ coor,
                           const float* __restrict__ pfeat,
                           const float* __restrict__ bev,
                           const float* __restrict__ W1,
                           const float* __restrict__ scale,
                           const float* __restrict__ shift,
                           const float* __restrict__ W2,
                           const float* __restrict__ b2,
                           float* __restrict__ out) {
  extern __shared__ char smem_raw[];
  float* xs      = (float*)smem_raw;
  float* w1s     = xs + ROWS_PER_BLOCK * XS_STRIDE;
  int*   bevbase = (int*)(w1s + IN1_DIM * XS_STRIDE);
  float* hs      = (float*)(bevbase + ROWS_PER_BLOCK);  // 64 x 68 normalized h
  float* w2s     = hs + ROWS_PER_BLOCK * HS_STRIDE;     // 32 x 68
  float* b2s     = w2s + NC_PAD * HS_STRIDE;            // 32

  const int tid = threadIdx.x;
  const int row0 = blockIdx.x * ROWS_PER_BLOCK;
  stage_w2(W2, b2, w2s, b2s);  // before stage_tiles' barriers
  stage_tiles(coor, pfeat, bev, W1, xs, w1s, bevbase, row0);

  const int wave = tid >> 5, lane = tid & 31;
  const int kh = lane >> 4, m = lane & 15;
  const float* xs_wave = xs + wave * 16 * XS_STRIDE;

  v8f acc[4];
#pragma unroll
  for (int t = 0; t < 4; ++t) acc[t] = v8f_zero();
  compute_h_tile(xs_wave, w1s, m, kh, acc);

  // Normalize + ReLU in D layout, spill to wave-private LDS rows (same-wave
  // DS ops are in-order, no barrier needed).
  const int lrow = wave * 16 + kh * 8;
#pragma unroll
  for (int t = 0; t < 4; ++t) {
    const float sc = scale[t * 16 + m];
    const float sh = shift[t * 16 + m];
#pragma unroll
    for (int v = 0; v < 8; ++v) {
      float x = fmaxf(acc[t][v] * sc + sh, 0.0f);
      hs[(lrow + v) * HS_STRIDE + t * 16 + m] = x;
    }
  }

  const float* hrow = hs + (wave * 16 + m) * HS_STRIDE;
  gemm2_lds<false>(hrow, w2s, nullptr, nullptr, b2s, out, row0 + wave * 16, m, kh);
}

// ---------------------------------------------------------------------------
extern "C" void kernel_launch(void* const* d_in, const int* in_sizes, int n_in,
                              void* d_out, int out_size, void* d_ws, size_t ws_size,
                              hipStream_t stream) {
  const int*   coor  = (const int*)d_in[0];
  const float* pfeat = (const float*)d_in[1];
  const float* bev   = (const float*)d_in[2];
  const float* voxel = (const float*)d_in[3];
  const float* W1    = (const float*)d_in[4];
  // d_in[5] = b1 : provably cancels inside BatchNorm, unused.
  const float* gamma = (const float*)d_in[6];
  const float* beta  = (const float*)d_in[7];
  const float* W2    = (const float*)d_in[8];
  const float* b2    = (const float*)d_in[9];
  float* out = (float*)d_out;

  const size_t hbytes = (size_t)NPTS * IN1_DIM * sizeof(float);
  const bool useH = ws_size >= hbytes + 1024;
  float* hws   = (float*)d_ws;
  float* stats = useH ? (float*)((char*)d_ws + hbytes) : (float*)d_ws;
  float* gsum = stats, *gsumsq = stats + 64, *scale = stats + 128, *shift = stats + 192;

  hipMemsetAsync(stats, 0, 128 * sizeof(float), stream);

  const int nblocks = (NPTS + ROWS_PER_BLOCK - 1) / ROWS_PER_BLOCK;
  const size_t smem1 =
      (2 * ROWS_PER_BLOCK * XS_STRIDE + ROWS_PER_BLOCK + 2 * IN1_DIM) * sizeof(float);
  const size_t smem2a =
      (ROWS_PER_BLOCK * HS_STRIDE + NC_PAD * HS_STRIDE + 2 * IN1_DIM + NC_PAD) *
      sizeof(float);
  const size_t smem2b =
      (2 * ROWS_PER_BLOCK * XS_STRIDE + ROWS_PER_BLOCK +
       ROWS_PER_BLOCK * HS_STRIDE + NC_PAD * HS_STRIDE + NC_PAD) * sizeof(float);

  if (useH) {
    vpr_gemm1_stats_kernel<true><<<nblocks, THREADS, smem1, stream>>>(
        coor, pfeat, bev, W1, hws, gsum, gsumsq);
  } else {
    vpr_gemm1_stats_kernel<false><<<nblocks, THREADS, smem1, stream>>>(
        coor, pfeat, bev, W1, nullptr, gsum, gsumsq);
  }

  vpr_bn_finalize_kernel<<<1, 64, 0, stream>>>(gsum, gsumsq, gamma, beta, scale, shift);

  if (useH) {
    vpr_gemm2_from_h_kernel<<<nblocks, THREADS, smem2a, stream>>>(
        hws, scale, shift, W2, b2, out);
  } else {
    vpr_gemm2_recompute_kernel<<<nblocks, THREADS, smem2b, stream>>>(
        coor, pfeat, bev, W1, scale, shift, W2, b2, out);
  }

  // voxel_preds is a pass-through copy.
  hipMemcpyAsync(out + (size_t)NPTS * NC, voxel,
                 (size_t)4 * NC * HW * HW * sizeof(float),
                 hipMemcpyDeviceToDevice, stream);
}